// HyperbolicPromptPool_59794534695467
// MI455X (gfx1250) — compile-verified
//
#include <hip/hip_runtime.h>
#include <hip/hip_bf16.h>
#include <math.h>

// ---------------- problem constants ----------------
#define B       128
#define T       512
#define D       768
#define P       60
#define PP      64          // pool padded to 64 for WMMA N tiles
#define TOPK    5
#define LENGTH  5
#define HEAD    (TOPK*LENGTH)     // 25
#define ROWS    (HEAD + T)        // 537
#define MAP_SCALE 0.1f
#define MAX_NORM  (1.0f - 4e-3f)

typedef float v2f __attribute__((ext_vector_type(2)));
typedef float v8f __attribute__((ext_vector_type(8)));

// ---------------- block reduction over 768 threads ----------------
__device__ __forceinline__ float block_sum_768(float v, float* s) {
    const int tid = threadIdx.x;
    s[tid] = v;
    __syncthreads();
    if (tid < 256) s[tid] += s[tid + 256] + s[tid + 512];
    __syncthreads();
    for (int off = 128; off >= 1; off >>= 1) {
        if (tid < off) s[tid] += s[tid + off];
        __syncthreads();
    }
    float r = s[0];
    __syncthreads();
    return r;
}

// ---------------- kernel 1: fused mean-partials + tail copy ----------------
// grid (B, 4), block 192 threads; thread owns one float4 column.
__global__ void __launch_bounds__(192)
k_mean_copy(const float* __restrict__ x, float* __restrict__ out0,
            float* __restrict__ partial) {
    const int b     = blockIdx.x;
    const int chunk = blockIdx.y;           // 4 chunks of 128 rows
    const int col   = threadIdx.x * 4;
    const float* src = x    + (size_t)b * T * D;
    float*       dst = out0 + (size_t)b * ROWS * D + (size_t)HEAD * D;
    float4 acc = make_float4(0.f, 0.f, 0.f, 0.f);
    const int t0 = chunk * (T / 4);
    for (int r = 0; r < T / 4; ++r) {
        const int t = t0 + r;
        float4 v = *reinterpret_cast<const float4*>(src + (size_t)t * D + col);
        *reinterpret_cast<float4*>(dst + (size_t)t * D + col) = v;
        acc.x += v.x; acc.y += v.y; acc.z += v.z; acc.w += v.w;
    }
    *reinterpret_cast<float4*>(partial + ((size_t)(b * 4 + chunk)) * D + col) = acc;
}

// ---------------- map_to_ball core (per thread owns one d) ----------------
__device__ __forceinline__ void map_to_ball(float xin, float* s,
                                            float* y_out, float* yn2_out) {
    float ss  = block_sum_768(xin * xin, s);
    float inv = rsqrtf(fmaxf(ss, 1e-12f));
    float u   = xin * inv * MAP_SCALE;
    float su  = block_sum_768(u * u, s);
    float n   = sqrtf(fmaxf(su, 1e-15f));
    float fac = tanhf(n) / n;
    float y   = u * fac;
    float yn2 = su * fac * fac;
    float ny  = sqrtf(fmaxf(yn2, 1e-15f));
    if (ny > MAX_NORM) {
        float sc = MAX_NORM / ny;
        y   *= sc;
        yn2 *= sc * sc;
    }
    *y_out = y; *yn2_out = yn2;
}

// ---------------- kernel 2: finish mean, map query to ball ----------------
// grid B, block 768
__global__ void __launch_bounds__(768)
k_map_q(const float* __restrict__ partial, float* __restrict__ qball,
        float* __restrict__ x2) {
    __shared__ float s[768];
    const int b = blockIdx.x, d = threadIdx.x;
    const float* p = partial + (size_t)b * 4 * D + d;
    float xm = (p[0] + p[D] + p[2 * D] + p[3 * D]) * (1.0f / (float)T);
    float y, yn2;
    map_to_ball(xm, s, &y, &yn2);
    qball[(size_t)b * D + d] = y;
    if (d == 0) x2[b] = yn2;
}

// ---------------- kernel 3: map keys to ball (zero-padded to PP) ----------
// grid PP, block 768
__global__ void __launch_bounds__(768)
k_map_k(const float* __restrict__ pkey, float* __restrict__ kball,
        float* __restrict__ y2) {
    __shared__ float s[768];
    const int p = blockIdx.x, d = threadIdx.x;
    float xin = (p < P) ? pkey[(size_t)p * D + d] : 0.0f;
    float y, yn2;
    map_to_ball(xin, s, &y, &yn2);          // pad rows map to exactly 0
    kball[(size_t)p * D + d] = y;
    if (d == 0) y2[p] = yn2;
}

// ---------------- kernel 4: WMMA f32 GEMM + Poincare epilogue -------------
// grid (PP/16, B/16), block 32 (one wave per 16x16 tile)
__global__ void __launch_bounds__(32)
k_dist(const float* __restrict__ qball, const float* __restrict__ kball,
       const float* __restrict__ x2, const float* __restrict__ y2,
       float* __restrict__ distw, float* __restrict__ sim) {
    const int n0   = blockIdx.x * 16;
    const int m0   = blockIdx.y * 16;
    const int lane = threadIdx.x;
    const int lm   = lane & 15;
    const int hi   = lane >> 4;
    const float* qrow = qball + (size_t)(m0 + lm) * D;   // A row (M)
    const float* krow = kball + (size_t)(n0 + lm) * D;   // B col (N) = key row
    v8f acc = {};
    for (int k = 0; k < D; k += 4) {
        // A 16x4 f32: lane=M, vgpr j -> K = 2*hi + j  (contiguous pair)
        v2f a = *reinterpret_cast<const v2f*>(qrow + k + 2 * hi);
        // B 4x16 f32: lane=N, vgpr j -> K = 2*hi + j
        v2f bf = *reinterpret_cast<const v2f*>(krow + k + 2 * hi);
        acc = __builtin_amdgcn_wmma_f32_16x16x4_f32(
            false, a, false, bf, (short)0, acc, false, false);
    }
    float fa[8];
    __builtin_memcpy(fa, &acc, sizeof(fa));
    const int   n  = n0 + lm;
    const float Y2 = y2[n];
#pragma unroll
    for (int v = 0; v < 8; ++v) {
        const int   m  = m0 + v + 8 * hi;        // C/D layout: M = v + 8*hi
        const float xy = fa[v];
        const float X2 = x2[m];
        const float a1  = 1.0f - 2.0f * xy + Y2;     // coeff of (-x)
        const float b1  = 1.0f - X2;                 // coeff of y
        const float den = fmaxf(1.0f - 2.0f * xy + X2 * Y2, 1e-15f);
        float dd  = (a1 * a1 * X2 + b1 * b1 * Y2 - 2.0f * a1 * b1 * xy) / (den * den);
        float nr  = sqrtf(fmaxf(dd, 1e-15f));
        nr = fminf(nr, 1.0f - 1e-7f);
        const float dist = logf((1.0f + nr) / (1.0f - nr));   // 2*artanh(nr)
        distw[m * PP + n] = dist;
        if (n < P) sim[m * P + n] = -dist;
    }
}

// ---------------- kernel 5: top-k, sort, reduce_sim ----------------------
// single block of B threads
__global__ void __launch_bounds__(B)
k_topk(const float* __restrict__ distw, float* __restrict__ out_idx_f,
       int* __restrict__ idx_i, float* __restrict__ reduce_sim) {
    __shared__ float s[B];
    const int b = threadIdx.x;
    const float* drow = distw + b * PP;
    int chosen[TOPK];
#pragma unroll
    for (int k = 0; k < TOPK; ++k) {
        float best = -1e30f; int bi = 0;
        for (int p = 0; p < P; ++p) {
            bool taken = false;
            for (int j = 0; j < k; ++j) taken |= (chosen[j] == p);
            if (taken) continue;
            float sv = -drow[p];                 // similarity
            if (sv > best) { best = sv; bi = p; }   // strict > == lowest-idx ties
        }
        chosen[k] = bi;
    }
    // insertion sort ascending
    for (int i = 1; i < TOPK; ++i) {
        int key = chosen[i], j = i - 1;
        while (j >= 0 && chosen[j] > key) { chosen[j + 1] = chosen[j]; --j; }
        chosen[j + 1] = key;
    }
    float sd = 0.f;
    for (int k = 0; k < TOPK; ++k) {
        out_idx_f[b * TOPK + k] = (float)chosen[k];
        idx_i[b * TOPK + k]     = chosen[k];
        sd += drow[chosen[k]];
    }
    s[b] = sd;
    __syncthreads();
    for (int off = B / 2; off >= 1; off >>= 1) {
        if (b < off) s[b] += s[b + off];
        __syncthreads();
    }
    if (b == 0) *reduce_sim = s[0] / (float)B;
}

// ---------------- kernel 6: gather prompts + selected keys ---------------
// grid (B, TOPK), block 768
__global__ void __launch_bounds__(768)
k_gather(const float* __restrict__ prompt, const float* __restrict__ kball,
         const int* __restrict__ idx_i, float* __restrict__ out0,
         float* __restrict__ out_selkey) {
    const int b = blockIdx.x, k = blockIdx.y, d = threadIdx.x;
    const int pi = idx_i[b * TOPK + k];
    const float* psrc = prompt + (size_t)pi * LENGTH * D;
    float*       pdst = out0   + (size_t)b * ROWS * D + (size_t)(k * LENGTH) * D;
#pragma unroll
    for (int r = 0; r < LENGTH; ++r)
        pdst[(size_t)r * D + d] = psrc[(size_t)r * D + d];
    out_selkey[(size_t)(b * TOPK + k) * D + d] = kball[(size_t)pi * D + d];
}

// ---------------- launcher ----------------
extern "C" void kernel_launch(void* const* d_in, const int* in_sizes, int n_in,
                              void* d_out, int out_size, void* d_ws, size_t ws_size,
                              hipStream_t stream) {
    const float* x      = (const float*)d_in[0];   // [B,T,D]
    const float* prompt = (const float*)d_in[1];   // [P,LENGTH,D]
    const float* pkey   = (const float*)d_in[2];   // [P,D]

    // outputs concatenated flat in return order
    float* out      = (float*)d_out;
    float* out0     = out;                                   // [B,ROWS,D]
    float* out_rsim = out0 + (size_t)B * ROWS * D;           // [1]
    float* out_idx  = out_rsim + 1;                          // [B,TOPK] as float
    float* out_sim  = out_idx + (size_t)B * TOPK;            // [B,P]
    float* out_sel  = out_sim + (size_t)B * P;               // [B,TOPK,D]

    // workspace layout (floats)
    float* ws        = (float*)d_ws;
    float* w_partial = ws;                                   // B*4*D
    float* w_qball   = w_partial + (size_t)B * 4 * D;        // B*D
    float* w_x2      = w_qball + (size_t)B * D;              // B
    float* w_kball   = w_x2 + B;                             // PP*D
    float* w_y2      = w_kball + (size_t)PP * D;             // PP
    float* w_dist    = w_y2 + PP;                            // B*PP
    int*   w_idx     = (int*)(w_dist + (size_t)B * PP);      // B*TOPK ints

    k_mean_copy<<<dim3(B, 4), 192, 0, stream>>>(x, out0, w_partial);
    k_map_q   <<<B, 768, 0, stream>>>(w_partial, w_qball, w_x2);
    k_map_k   <<<PP, 768, 0, stream>>>(pkey, w_kball, w_y2);
    k_dist    <<<dim3(PP / 16, B / 16), 32, 0, stream>>>(w_qball, w_kball,
                                                         w_x2, w_y2, w_dist, out_sim);
    k_topk    <<<1, B, 0, stream>>>(w_dist, out_idx, w_idx, out_rsim);
    k_gather  <<<dim3(B, TOPK), 768, 0, stream>>>(prompt, w_kball, w_idx,
                                                  out0, out_sel);
}